// Attention_24996709663156
// MI455X (gfx1250) — compile-verified
//
#include <hip/hip_runtime.h>
#include <math.h>
#include <stdint.h>

typedef __attribute__((ext_vector_type(2))) float v2f;
typedef __attribute__((ext_vector_type(8))) float v8f;

#define HDIM    512
#define SEQ     8192
#define BATCH   32
#define CHUNK   128                 /* rows per block; whole chunk lives in LDS */
#define NCHUNK  (SEQ / CHUNK)       /* 64 */
#define REC_SZ  (2 + HDIM)          /* m_c, l_c, c_c[512] */
#define ESTRIDE 516                 /* padded LDS row stride: 516%64==4 -> no bank conflicts */

/* LDS budget (floats): tile 128*516 + states 512 + la 128 + p 128 + red 128 */
#define LDS_FLOATS (CHUNK * ESTRIDE + HDIM + CHUNK + CHUNK + CHUNK)
#define LDS_BYTES  (LDS_FLOATS * 4) /* ~261 KB < 320 KB */

// ---------------------------------------------------------------------------
// Pass 1: one (batch, 128-row chunk) per block, 8 waves.
//   Per wave (no block barrier until context phase):
//     - redundantly stage states_h[b] into LDS (same values, benign)
//     - GLOBAL_LOAD_ASYNC_TO_LDS_B128 its own 16-row tile (32 KB, coalesced)
//     - s_wait_asynccnt 0 (own tile only)  -> WMMA f32 16x16x4 scores
//   Then barrier; chunk softmax stats + chunk-partial context from LDS.
//   enc_out crosses HBM exactly once, fully coalesced.
// ---------------------------------------------------------------------------
__global__ __launch_bounds__(256) void attn_pass1(
    const float* __restrict__ states,   // (B, 512)
    const float* __restrict__ enc,      // (B, 8192, 512)
    float* __restrict__ wsAlign,        // (B, 8192)
    float* __restrict__ wsRec)          // (B, NCHUNK, REC_SZ)
{
  extern __shared__ float smem[];
  float* tileE = smem;                       // CHUNK x ESTRIDE
  float* st    = tileE + CHUNK * ESTRIDE;    // HDIM
  float* la    = st + HDIM;                  // CHUNK
  float* pbuf  = la + CHUNK;                 // CHUNK
  float* red   = pbuf + CHUNK;               // CHUNK

  const int t     = threadIdx.x;
  const int b     = blockIdx.x / NCHUNK;
  const int chunk = blockIdx.x % NCHUNK;
  const int s0    = chunk * CHUNK;
  const float* Eb = enc + (size_t)b * SEQ * HDIM;

  const int wave  = t >> 5;
  const int lane  = t & 31;
  const int rbase = wave * 16;          // wave-owned rows [rbase, rbase+16)

  // ---- every wave stages states_h[b] (same values -> benign race) ----
  for (int i = lane; i < HDIM; i += 32) st[i] = states[b * HDIM + i];

  // ---- wave async-stages its own 16x512 tile: 64 instrs x 32 lanes x 16B ----
  {
    const uint64_t gbase = (uint64_t)(uintptr_t)(Eb + (size_t)s0 * HDIM);
    const uint32_t tbase = (uint32_t)(uintptr_t)tileE;   // LDS byte offset
    for (int r = 0; r < 64; ++r) {
      const int id  = r * 32 + lane;    // chunk id within wave tile
      const int row = rbase + (id >> 7);
      const int c16 = id & 127;         // 16B chunk index in row
      const uint32_t goff = (uint32_t)((row * HDIM    + c16 * 4) * 4);
      const uint32_t loff = tbase + (uint32_t)((row * ESTRIDE + c16 * 4) * 4);
      asm volatile("global_load_async_to_lds_b128 %0, %1, %2"
                   :: "v"(loff), "v"(goff), "s"(gbase) : "memory");
    }
    asm volatile("s_wait_asynccnt 0x0" ::: "memory");  // own tile landed
  }

  // ---- WMMA scores for the wave's own tile (no block barrier needed) ----
  {
    const int mrow  = lane & 15;          // M row inside the 16-row tile
    const int khalf = (lane >> 4) * 2;    // lanes 16-31 hold K=2,3
    const float* rowp = tileE + (size_t)(rbase + mrow) * ESTRIDE;
    v8f acc = {};
    for (int k = 0; k < HDIM; k += 4) {
      v2f a  = *(const v2f*)(rowp + k + khalf);        // A[M][K] = E[row][k+K]
      v2f bb = *(const v2f*)(st   + k + khalf);        // B[K][N] = states[k+K]
      acc = __builtin_amdgcn_wmma_f32_16x16x4_f32(
          false, a, false, bb, (short)0, acc, false, false);
    }
    // D: lanes 0-15 VGPR r -> M=r ; lanes 16-31 VGPR r -> M=8+r (cols identical)
    const int e = lane & 7;
    float v;
    switch (e) {
      case 0: v = acc[0]; break;
      case 1: v = acc[1]; break;
      case 2: v = acc[2]; break;
      case 3: v = acc[3]; break;
      case 4: v = acc[4]; break;
      case 5: v = acc[5]; break;
      case 6: v = acc[6]; break;
      default: v = acc[7]; break;
    }
    const bool ownA = (lane < 8);
    const bool ownB = (lane >= 16 && lane < 24);
    if (ownA || ownB) {
      const int lrow = rbase + (ownA ? lane : (lane - 8));
      const float al = v * (1.0f / (float)HDIM);
      la[lrow] = al;
      wsAlign[(size_t)b * SEQ + s0 + lrow] = al;
    }
  }
  __syncthreads();   // all tiles staged + all scores in la

  // ---- chunk max over 128 scores ----
  if (t < CHUNK) red[t] = la[t];
  __syncthreads();
  for (int off = CHUNK / 2; off > 0; off >>= 1) {
    if (t < off) red[t] = fmaxf(red[t], red[t + off]);
    __syncthreads();
  }
  const float m_c = red[0];
  __syncthreads();

  // ---- chunk exp-sum ----
  if (t < CHUNK) {
    const float p = __expf(la[t] - m_c);
    pbuf[t] = p;
    red[t]  = p;
  }
  __syncthreads();
  for (int off = CHUNK / 2; off > 0; off >>= 1) {
    if (t < off) red[t] += red[t + off];
    __syncthreads();
  }
  const float l_c = red[0];
  __syncthreads();

  // ---- chunk-partial context from LDS (h = t, t+256) ----
  float c0 = 0.f, c1 = 0.f;
  for (int s = 0; s < CHUNK; ++s) {
    const float ps  = pbuf[s];
    const float* rp = tileE + (size_t)s * ESTRIDE;
    c0 = fmaf(ps, rp[t],       c0);
    c1 = fmaf(ps, rp[t + 256], c1);
  }
  float* rec = wsRec + (size_t)(b * NCHUNK + chunk) * REC_SZ;
  if (t == 0) { rec[0] = m_c; rec[1] = l_c; }
  rec[2 + t]       = c0;
  rec[2 + t + 256] = c1;
}

// ---------------------------------------------------------------------------
// Pass 2: per-batch log-sum-exp combine + normalized weights
// ---------------------------------------------------------------------------
__global__ __launch_bounds__(256) void attn_pass2(
    const float* __restrict__ wsAlign,
    const float* __restrict__ wsRec,
    float* __restrict__ outCtx,     // (B, 1, 512)
    float* __restrict__ outW)       // (B, 8192, 1)
{
  __shared__ float mc[NCHUNK];
  __shared__ float lc[NCHUNK];
  __shared__ float m_s, L_s;

  const int t = threadIdx.x;
  const int b = blockIdx.x;
  const float* recs = wsRec + (size_t)b * NCHUNK * REC_SZ;

  if (t < NCHUNK) {
    mc[t] = recs[(size_t)t * REC_SZ + 0];
    lc[t] = recs[(size_t)t * REC_SZ + 1];
  }
  __syncthreads();

  if (t == 0) {
    float m = mc[0];
    for (int i = 1; i < NCHUNK; ++i) m = fmaxf(m, mc[i]);
    float L = 0.f;
    for (int i = 0; i < NCHUNK; ++i) L += lc[i] * __expf(mc[i] - m);
    m_s = m; L_s = L;
  }
  __syncthreads();
  const float m    = m_s;
  const float invL = 1.0f / L_s;

  float c0 = 0.f, c1 = 0.f;
  for (int i = 0; i < NCHUNK; ++i) {
    const float sc = __expf(mc[i] - m);
    c0 = fmaf(recs[(size_t)i * REC_SZ + 2 + t],       sc, c0);
    c1 = fmaf(recs[(size_t)i * REC_SZ + 2 + t + 256], sc, c1);
  }
  outCtx[(size_t)b * HDIM + t]       = c0 * invL;
  outCtx[(size_t)b * HDIM + t + 256] = c1 * invL;

  for (int s = t; s < SEQ; s += 256) {
    outW[(size_t)b * SEQ + s] =
        __expf(wsAlign[(size_t)b * SEQ + s] - m) * invL;
  }
}

// ---------------------------------------------------------------------------
extern "C" void kernel_launch(void* const* d_in, const int* in_sizes, int n_in,
                              void* d_out, int out_size, void* d_ws, size_t ws_size,
                              hipStream_t stream) {
  const float* states = (const float*)d_in[0];  // (32, 512)
  const float* enc    = (const float*)d_in[1];  // (32, 8192, 512)

  float* out    = (float*)d_out;
  float* outCtx = out;                          // 32*512 floats (context first)
  float* outW   = out + BATCH * HDIM;           // 32*8192 floats (weights)

  float* wsAlign = (float*)d_ws;                          // 1 MB
  float* wsRec   = wsAlign + (size_t)BATCH * SEQ;         // 32*64*514 floats (~4.2 MB)

  attn_pass1<<<BATCH * NCHUNK, 256, LDS_BYTES, stream>>>(states, enc, wsAlign, wsRec);
  attn_pass2<<<BATCH, 256, 0, stream>>>(wsAlign, wsRec, outCtx, outW);
}